// YOLOPredict_16003048145237
// MI455X (gfx1250) — compile-verified
//
#include <hip/hip_runtime.h>
#include <math.h>
#include <stdint.h>

// ---------------- problem constants ----------------
#define NA      524288      // anchors
#define NC      80          // classes
#define STRIDE  85          // 5 + NC
#define CONF    0.1f
#define TOPK    512
#define BINS    64
#define BIN_SCALE 2.0f      // bin = floor(score*2), covers [0, 32)
#define CAND    2048        // per-class candidate capacity after thresholding
#define ROWS_PER_TILE 128   // rows staged per TDM transfer

typedef float v2f __attribute__((ext_vector_type(2)));
typedef float v8f __attribute__((ext_vector_type(8)));
typedef unsigned int v4u __attribute__((ext_vector_type(4)));
typedef int v4i __attribute__((ext_vector_type(4)));
typedef int v8i __attribute__((ext_vector_type(8)));

// ---------------- workspace layout (units of 4 bytes) ----------------
#define WS_BOXES 0                          // float 4*NA   : decoded clipped boxes
#define WS_HIST  (4*NA)                     // int   NC*BINS
#define WS_THR   (WS_HIST + NC*BINS)        // float NC
#define WS_CNT   (WS_THR + NC)              // int   NC
#define WS_VALS  (WS_CNT + NC)              // float NC*TOPK (top-k scores pre-mask)
#define WS_CSC   (WS_VALS + NC*TOPK)        // float NC*CAND (candidate scores)
#define WS_CIX   (WS_CSC + NC*CAND)         // int   NC*CAND (candidate anchor ids)

// ---------------- output layout (floats) ----------------
#define OUT_BOX   0
#define OUT_SCORE (NC*TOPK*4)
#define OUT_LABEL (OUT_SCORE + NC*TOPK)
#define OUT_KEEP  (OUT_LABEL + NC*TOPK)

// ---- TDM: DMA a [ROWS_PER_TILE x STRIDE] f32 tile of pred into LDS ----
// D# per CDNA5 ISA: group0 {count=1, lds_addr,
// global_addr[56:0], type=2}; group1 {data_size=4B, tensor_dim0=85,
// tile_dim0=85, tile_dim1=ROWS, tensor_dim0_stride=85}; groups 2/3 = 0 (2-D).
__device__ __forceinline__ void tdm_load_rows(const float* gsrc, float* lds_dst) {
    const uint64_t ga = (uint64_t)(uintptr_t)gsrc;
    // generic LDS pointer: aperture in high 32 bits, byte offset in low 32
    const unsigned lds_off = (unsigned)(uintptr_t)(void*)lds_dst;
    v4u g0;
    g0.x = 1u;                                            // count=1, user D#
    g0.y = lds_off;                                       // lds_addr (bytes)
    g0.z = (unsigned)(ga & 0xFFFFFFFFu);                  // global_addr[31:0]
    g0.w = (unsigned)((ga >> 32) & 0x01FFFFFFu) | (2u << 30); // addr[56:32], type=2
    v8i g1;
    g1[0] = 0x00020000;                  // workgroup_mask=0, data_size=2 (4B)
    g1[1] = (STRIDE << 16);              // tensor_dim0[15:0] in bits[31:16]
    g1[2] = 0;                           // tensor_dim0 hi / tensor_dim1 lo
    g1[3] = (NA >> 16) | (STRIDE << 16); // tensor_dim1[31:16], tile_dim0=85
    g1[4] = ROWS_PER_TILE;               // tile_dim1, tile_dim2=0
    g1[5] = STRIDE;                      // tensor_dim0_stride[31:0]
    g1[6] = 0;
    g1[7] = 0;
    v4i gz = {0, 0, 0, 0};
#if defined(__clang_major__) && (__clang_major__ >= 23)
    v8i gz8 = {0, 0, 0, 0, 0, 0, 0, 0};
    __builtin_amdgcn_tensor_load_to_lds(g0, g1, gz, gz, gz8, 0);
#else
    __builtin_amdgcn_tensor_load_to_lds(g0, g1, gz, gz, 0);
#endif
    __builtin_amdgcn_s_wait_tensorcnt((short)0);
}

// ================= kernel 0: zero hist / counters =================
__global__ void k_zero(float* ws) {
    int i = blockIdx.x * blockDim.x + threadIdx.x;
    int* p = (int*)ws + WS_HIST;
    const int total = (WS_CNT + NC) - WS_HIST;   // hist + thr + cnt
    if (i < total) p[i] = 0;
}

// ======= kernel 1: TDM-staged decode + per-class score histogram =======
// 256 threads, 256 rows per block in two 128-row TDM tiles. Thread pair
// (tid>>1, tid&1) splits the 80 classes of one row.
__global__ void __launch_bounds__(256) k_decode_hist(const float* __restrict__ pred,
                                                     float* __restrict__ ws) {
    __shared__ float stage[ROWS_PER_TILE * STRIDE];   // 43,520 B
    __shared__ int h[NC * BINS];                      // 20,480 B
    const int tid = threadIdx.x;
    for (int i = tid; i < NC * BINS; i += 256) h[i] = 0;

    const int r    = tid >> 1;
    const int half = tid & 1;

    for (int chunk = 0; chunk < 2; ++chunk) {
        const int row0 = blockIdx.x * 256 + chunk * ROWS_PER_TILE;
        if (tid == 0)
            tdm_load_rows(pred + (size_t)row0 * STRIDE, stage);
        __syncthreads();                              // also orders h zero-init

        const float* row = stage + r * STRIDE;
        const float obj = row[4];
        if (half == 0) {
            const float cx = row[0], cy = row[1], w = row[2], hh = row[3];
            const float x1 = fminf(fmaxf(cx - w * 0.5f, 0.f), 1.f);
            const float y1 = fminf(fmaxf(cy - hh * 0.5f, 0.f), 1.f);
            const float x2 = fminf(fmaxf(cx + w * 0.5f, 0.f), 1.f);
            const float y2 = fminf(fmaxf(cy + hh * 0.5f, 0.f), 1.f);
            ((float4*)(ws + WS_BOXES))[row0 + r] = make_float4(x1, y1, x2, y2);
        }
        const int cbase = half * 40;
        #pragma unroll 4
        for (int c = 0; c < 40; ++c) {
            const float s = row[5 + cbase + c] * obj;
            if (s > CONF) {
                int bin = (int)(s * BIN_SCALE);
                bin = bin > (BINS - 1) ? (BINS - 1) : bin;
                atomicAdd(&h[(cbase + c) * BINS + bin], 1);
            }
        }
        __syncthreads();                              // stage reuse / final adds done
    }
    int* gh = (int*)ws + WS_HIST;
    for (int i = tid; i < NC * BINS; i += 256)
        if (h[i]) atomicAdd(&gh[i], h[i]);
}

// ======= kernel 2: per-class threshold from histogram =======
__global__ void k_thresh(float* ws) {
    const int c = threadIdx.x;
    if (c >= NC) return;
    const int* gh = (const int*)ws + WS_HIST;
    float t = CONF;
    int cum = 0;
    for (int b = BINS - 1; b >= 0; --b) {
        cum += gh[c * BINS + b];
        if (cum >= TOPK) { t = fmaxf(CONF, (float)b / BIN_SCALE); break; }
    }
    (ws + WS_THR)[c] = t;
}

// ======= kernel 3: TDM-staged candidate compaction =======
__global__ void __launch_bounds__(256) k_compact(const float* __restrict__ pred,
                                                 float* __restrict__ ws) {
    __shared__ float stage[ROWS_PER_TILE * STRIDE];   // 43,520 B
    __shared__ float t[NC];
    const int tid = threadIdx.x;
    if (tid < NC) t[tid] = (ws + WS_THR)[tid];

    int*   cnt = (int*)ws + WS_CNT;
    float* csc = ws + WS_CSC;
    int*   cix = (int*)ws + WS_CIX;

    const int r    = tid >> 1;
    const int half = tid & 1;

    for (int chunk = 0; chunk < 2; ++chunk) {
        const int row0 = blockIdx.x * 256 + chunk * ROWS_PER_TILE;
        if (tid == 0)
            tdm_load_rows(pred + (size_t)row0 * STRIDE, stage);
        __syncthreads();

        const int a = row0 + r;
        const float* row = stage + r * STRIDE;
        const float obj = row[4];
        const int cbase = half * 40;
        #pragma unroll 4
        for (int c = 0; c < 40; ++c) {
            const int cc = cbase + c;
            const float s = row[5 + cc] * obj;
            if (s > CONF && s >= t[cc]) {
                const int p = atomicAdd(&cnt[cc], 1);
                if (p < CAND) {
                    csc[cc * CAND + p] = s;
                    cix[cc * CAND + p] = a;
                }
            }
        }
        __syncthreads();
    }
}

// ======= kernel 4: per-class exact top-512 via LDS bitonic sort =======
__global__ void __launch_bounds__(256) k_topk(float* __restrict__ ws,
                                              float* __restrict__ out) {
    __shared__ float ss[CAND];
    __shared__ int   ii[CAND];
    const int c = blockIdx.x, tid = threadIdx.x;

    int n = ((int*)ws + WS_CNT)[c];
    n = n > CAND ? CAND : n;
    const float* csc = ws + WS_CSC + c * CAND;
    const int*   cix = (const int*)ws + WS_CIX + c * CAND;

    for (int i = tid; i < CAND; i += blockDim.x) {
        if (i < n) { ss[i] = csc[i]; ii[i] = cix[i]; }
        else       { ss[i] = -INFINITY; ii[i] = -1; }
    }
    __syncthreads();

    for (int k = 2; k <= CAND; k <<= 1) {
        for (int j = k >> 1; j > 0; j >>= 1) {
            for (int i = tid; i < CAND; i += blockDim.x) {
                const int ixj = i ^ j;
                if (ixj > i) {
                    const bool desc = ((i & k) == 0);
                    const float si = ss[i], sj = ss[ixj];
                    if (desc ? (si < sj) : (si > sj)) {
                        ss[i] = sj; ss[ixj] = si;
                        const int tt = ii[i]; ii[i] = ii[ixj]; ii[ixj] = tt;
                    }
                }
            }
            __syncthreads();
        }
    }

    float* valsw = ws + WS_VALS + c * TOPK;
    const float4* boxes = (const float4*)(ws + WS_BOXES);
    for (int j = tid; j < TOPK; j += blockDim.x) {
        const float v = ss[j];
        const int   id = ii[j];
        valsw[j] = v;
        float4 b = make_float4(0.f, 0.f, 0.f, 0.f);
        if (id >= 0) b = boxes[id];
        const int o = (c * TOPK + j) * 4;
        out[OUT_BOX + o + 0] = b.x;
        out[OUT_BOX + o + 1] = b.y;
        out[OUT_BOX + o + 2] = b.z;
        out[OUT_BOX + o + 3] = b.w;
        out[OUT_LABEL + c * TOPK + j] = (float)c;
    }
}

// ======= kernel 5: per-class NMS (WMMA area-sum + ballot bitmask) =======
// iou > 0.5  <=>  2*inter > union = areasum - inter  <=>  3*inter > areasum,
// with areasum[m,n] = area_m + area_n computed by V_WMMA_F32_16X16X4_F32.
__global__ void __launch_bounds__(256) k_nms(float* __restrict__ ws,
                                             float* __restrict__ out) {
    __shared__ float bx[TOPK * 4];
    __shared__ float area[TOPK];
    __shared__ float vals[TOPK];
    __shared__ unsigned short sup[TOPK * 32];    // 512 x 512 bits
    __shared__ unsigned short keepA[32];

    const int c = blockIdx.x, tid = threadIdx.x;
    const float* vsrc = ws + WS_VALS + c * TOPK;

    for (int j = tid; j < TOPK; j += blockDim.x) {
        const int o = (c * TOPK + j) * 4;
        const float b0 = out[OUT_BOX + o + 0];
        const float b1 = out[OUT_BOX + o + 1];
        const float b2 = out[OUT_BOX + o + 2];
        const float b3 = out[OUT_BOX + o + 3];
        bx[4 * j + 0] = b0; bx[4 * j + 1] = b1;
        bx[4 * j + 2] = b2; bx[4 * j + 3] = b3;
        area[j] = fmaxf(b2 - b0, 0.f) * fmaxf(b3 - b1, 0.f);
        vals[j] = vsrc[j];
    }
    // zero suppression mask (strictly-lower tiles are skipped below)
    int* sup32 = (int*)sup;
    for (int i = tid; i < TOPK * 16; i += blockDim.x) sup32[i] = 0;
    __syncthreads();

    const int lane = tid & 31;
    const int wv   = tid >> 5;                   // 8 waves
    const int half = lane >> 4;
    const int l16  = lane & 15;

    for (int t = wv; t < 32 * 32; t += 8) {      // t is wave-uniform
        const int ti = t >> 5;
        const int tj = t & 31;
        if (tj < ti) continue;                   // n>m impossible: tile all-zero
        const int m0 = ti << 4;
        const int n0 = tj << 4;

        v2f A, B;
        A.x = half ? 0.f : area[m0 + l16];       // A[m,:]=[area_m,1,0,0]
        A.y = half ? 0.f : 1.f;
        B.x = half ? 0.f : 1.f;                  // B[:,n]=[1,area_n,0,0]
        B.y = half ? 0.f : area[n0 + l16];
        v8f Cz = {};
        v8f D = __builtin_amdgcn_wmma_f32_16x16x4_f32(
            false, A, false, B, (short)0, Cz, false, false);

        const int n = n0 + l16;
        const float nx1 = bx[4 * n + 0], ny1 = bx[4 * n + 1];
        const float nx2 = bx[4 * n + 2], ny2 = bx[4 * n + 3];

        #pragma unroll
        for (int rr = 0; rr < 8; ++rr) {
            const int m = m0 + rr + (half << 3);
            const float ix1 = fmaxf(bx[4 * m + 0], nx1);
            const float iy1 = fmaxf(bx[4 * m + 1], ny1);
            const float ix2 = fminf(bx[4 * m + 2], nx2);
            const float iy2 = fminf(bx[4 * m + 3], ny2);
            const float inter = fmaxf(ix2 - ix1, 0.f) * fmaxf(iy2 - iy1, 0.f);
            const bool s = (n > m) && (3.0f * inter > D[rr]);
            const unsigned int b32 = (unsigned int)__ballot(s);
            if (lane == 0)  sup[(m0 + rr) * 32 + tj]     = (unsigned short)(b32 & 0xFFFFu);
            if (lane == 16) sup[(m0 + 8 + rr) * 32 + tj] = (unsigned short)(b32 >> 16);
        }
    }
    __syncthreads();

    // sequential greedy keep: wave 0, lane j owns 16-bit column j
    if (tid < 32) {
        unsigned int kp = 0;
        #pragma unroll
        for (int b = 0; b < 16; ++b)
            if (vals[tid * 16 + b] > CONF) kp |= (1u << b);
        for (int i = 0; i < TOPK; ++i) {
            const unsigned int kv = (unsigned int)__shfl((int)kp, i >> 4, 32);
            if ((kv >> (i & 15)) & 1u)
                kp &= ~(unsigned int)sup[i * 32 + tid];
        }
        keepA[tid] = (unsigned short)kp;
    }
    __syncthreads();

    for (int j = tid; j < TOPK; j += blockDim.x) {
        const int bit = (keepA[j >> 4] >> (j & 15)) & 1;
        out[OUT_KEEP  + c * TOPK + j] = (float)bit;
        out[OUT_SCORE + c * TOPK + j] = bit ? vals[j] : 0.f;
    }
}

// ================= host launcher =================
extern "C" void kernel_launch(void* const* d_in, const int* in_sizes, int n_in,
                              void* d_out, int out_size, void* d_ws, size_t ws_size,
                              hipStream_t stream) {
    const float* pred = (const float*)d_in[0];
    float* out = (float*)d_out;
    float* ws  = (float*)d_ws;

    const int zeroN = (WS_CNT + NC) - WS_HIST;
    k_zero<<<(zeroN + 255) / 256, 256, 0, stream>>>(ws);
    k_decode_hist<<<NA / 256, 256, 0, stream>>>(pred, ws);
    k_thresh<<<1, 128, 0, stream>>>(ws);
    k_compact<<<NA / 256, 256, 0, stream>>>(pred, ws);
    k_topk<<<NC, 256, 0, stream>>>(ws, out);
    k_nms<<<NC, 256, 0, stream>>>(ws, out);
}